// ProsodyExtractor_26414048870853
// MI455X (gfx1250) — compile-verified
//
#include <hip/hip_runtime.h>
#include <math.h>

// ---------------------------------------------------------------------------
// ProsodyExtractor for MI455X (gfx1250, wave32, WMMA)
//   B=16, T=4096, E=256, K=256, P=128, STAT=515
// Pipeline:
//   1) pad W1 (515x256) -> W1p (528x256), zero rows 515..527
//   2) per-slot feature extraction -> featsP (4096 x 528), cols 515..527 = 0
//      (with global_prefetch of the next evidence row)
//   3) h = gelu(featsP @ W1p + b1)   via V_WMMA_F32_16X16X4_F32, 16x64/wave
//   4) y = h @ W2 + b2               via V_WMMA_F32_16X16X4_F32, 16x64/wave
//   5) out = LayerNorm(y) * ln_g + ln_b
// ---------------------------------------------------------------------------

typedef __attribute__((ext_vector_type(2))) float v2f;
typedef __attribute__((ext_vector_type(8))) float v8f;

constexpr int Bsz  = 16;
constexpr int Tsz  = 4096;
constexpr int Esz  = 256;
constexpr int Ksz  = 256;
constexpr int Psz  = 128;
constexpr int STAT = 2 * Esz + 3;   // 515
constexpr int KP   = 528;           // STAT padded to multiple of 16
constexpr int N1   = 2 * Psz;       // 256
constexpr int ROWS = Bsz * Ksz;     // 4096

// ---------------------------------------------------------------------------
// Kernel 1: zero-pad W1 (STAT x N1) into W1p (KP x N1)
// ---------------------------------------------------------------------------
__global__ void pe_pad_w1(const float* __restrict__ W1, float* __restrict__ W1p) {
  const int k = blockIdx.x;     // 0..KP-1
  const int n = threadIdx.x;    // 0..N1-1
  W1p[(size_t)k * N1 + n] = (k < STAT) ? W1[(size_t)k * N1 + n] : 0.0f;
}

// ---------------------------------------------------------------------------
// Kernel 2: per-slot statistics -> featsP row (KP floats, padded with zeros)
// grid = ROWS blocks (one per (b,k) slot), block = Esz = 256 threads.
// Thread tid owns evidence channel e = tid. Row-norm (energy) via block
// reduction per timestep; slot length <= 32. Next row prefetched to overlap
// HBM latency with the reduction (emits global_prefetch_b8).
// ---------------------------------------------------------------------------
__global__ void pe_feats(const float* __restrict__ evidence,
                         const long long* __restrict__ boundaries,  // int64 (B,K,2)
                         const int* __restrict__ slot_mask,
                         float* __restrict__ featsP) {
  const int row = blockIdx.x;          // b*K + k
  const int b   = row / Ksz;
  const int tid = threadIdx.x;         // channel index 0..255
  __shared__ float red[256];

  const int start = (int)boundaries[(size_t)row * 2 + 0];
  const int end   = (int)boundaries[(size_t)row * 2 + 1];
  const int mask  = slot_mask[row];

  const float* ev = evidence + (size_t)b * Tsz * Esz + tid;

  // energy just before the slot (reference: shifted[0] = energy[0] -> delta 0)
  const int tprev = (start > 0) ? (start - 1) : start;
  {
    float xp = ev[(size_t)tprev * Esz];
    __builtin_prefetch(ev + (size_t)start * Esz, 0, 3);  // speculative, safe
    red[tid] = xp * xp;
  }
  __syncthreads();
  for (int s = 128; s > 0; s >>= 1) {
    if (tid < s) red[tid] += red[tid + s];
    __syncthreads();
  }
  float prev = sqrtf(red[0]);
  __syncthreads();

  float sum = 0.f, sumsq = 0.f, e_sum = 0.f, d_sum = 0.f;
  for (int t = start; t < end; ++t) {
    float x = ev[(size_t)t * Esz];
    __builtin_prefetch(ev + (size_t)(t + 1) * Esz, 0, 3);  // next row
    sum   += x;
    sumsq += x * x;
    red[tid] = x * x;
    __syncthreads();
    for (int s = 128; s > 0; s >>= 1) {
      if (tid < s) red[tid] += red[tid + s];
      __syncthreads();
    }
    float en = sqrtf(red[0]);   // ||evidence[b,t,:]|| broadcast to all threads
    __syncthreads();
    e_sum += en;
    d_sum += fabsf(en - prev);
    prev = en;
  }

  if (!mask) { sum = 0.f; sumsq = 0.f; e_sum = 0.f; d_sum = 0.f; }
  const float cnt_raw = mask ? (float)(end - start) : 0.0f;
  const float cnt = fmaxf(cnt_raw, 1.0f);   // jnp.clip(counts, a_min=1)
  const float inv = 1.0f / cnt;
  const float mean = sum * inv;
  const float msq  = sumsq * inv;
  const float stdv = sqrtf(fmaxf(msq - mean * mean, 0.0f));

  float* frow = featsP + (size_t)row * KP;
  frow[tid]        = mean;
  frow[Esz + tid]  = stdv;
  if (tid == 0) {
    frow[2 * Esz + 0] = cnt * (1.0f / 50.0f);  // duration
    frow[2 * Esz + 1] = e_sum * inv;           // slot_energy
    frow[2 * Esz + 2] = d_sum * inv;           // slot_delta
  }
  if (tid < KP - STAT) frow[STAT + tid] = 0.0f;  // zero K padding
}

// ---------------------------------------------------------------------------
// FP32 WMMA GEMM: each wave computes a 16x64 strip (4 x 16x16 tiles) so the
// A fragment is reused across 4 WMMAs per k-step.
// A-frag (16x4 f32): lanes 0-15 -> rows M=0..15, v[0]=K0 v[1]=K1;
//                    lanes 16-31 -> v[0]=K2 v[1]=K3.
// B-frag (4x16 f32): VGPR0: lanes 0-15 K=0, lanes 16-31 K=2;
//                    VGPR1: lanes 0-15 K=1, lanes 16-31 K=3.  N = lane&15.
// C/D (16x16 f32):   VGPR j: lanes 0-15 -> M=j; lanes 16-31 -> M=j+8.
// ---------------------------------------------------------------------------

// Kernel 3: h = gelu(featsP(ROWS x KP) @ W1p(KP x N1) + b1)
// tiles = (ROWS/16) * (N1/64) = 1024 waves -> 128 blocks x 8 waves.
__global__ void pe_gemm1_gelu(const float* __restrict__ featsP,
                              const float* __restrict__ W1p,
                              const float* __restrict__ b1,
                              float* __restrict__ h) {
  const int tid  = threadIdx.x;
  const int wave = tid >> 5;
  const int lane = tid & 31;
  const int half = lane >> 4;
  const int l16  = lane & 15;
  const int tile = blockIdx.x * 8 + wave;   // 1024 strips
  const int row0 = (tile >> 2) * 16;        // 256 row tiles
  const int n0   = (tile & 3) * 64;         // 4 col groups of 64

  v8f acc[4] = {};
  const float* ap = featsP + (size_t)(row0 + l16) * KP + 2 * half;
  const float* bp = W1p + (size_t)(2 * half) * N1 + n0 + l16;
#pragma unroll 2
  for (int k0 = 0; k0 < KP; k0 += 4) {
    v2f a;
    a.x = ap[0];
    a.y = ap[1];
#pragma unroll
    for (int g = 0; g < 4; ++g) {
      v2f bb;
      bb.x = bp[g * 16];
      bb.y = bp[N1 + g * 16];
      acc[g] = __builtin_amdgcn_wmma_f32_16x16x4_f32(
          /*neg_a=*/false, a, /*neg_b=*/false, bb,
          /*c_mod=*/(short)0, acc[g], /*reuse_a=*/false, /*reuse_b=*/false);
    }
    ap += 4;
    bp += 4 * N1;
  }
#pragma unroll
  for (int g = 0; g < 4; ++g) {
#pragma unroll
    for (int j = 0; j < 8; ++j) {
      const int m = row0 + j + 8 * half;
      const int n = n0 + g * 16 + l16;
      float x = acc[g][j] + b1[n];
      // exact GELU (approximate=False): 0.5*x*(1+erf(x/sqrt(2)))
      float gl = 0.5f * x * (1.0f + erff(x * 0.70710678118654752440f));
      h[(size_t)m * N1 + n] = gl;
    }
  }
}

// Kernel 4: y = h(ROWS x N1) @ W2(N1 x Psz) + b2
// tiles = (ROWS/16) * (Psz/64) = 512 waves -> 64 blocks x 8 waves.
__global__ void pe_gemm2(const float* __restrict__ h,
                         const float* __restrict__ W2,
                         const float* __restrict__ b2,
                         float* __restrict__ y) {
  const int tid  = threadIdx.x;
  const int wave = tid >> 5;
  const int lane = tid & 31;
  const int half = lane >> 4;
  const int l16  = lane & 15;
  const int tile = blockIdx.x * 8 + wave;   // 512 strips
  const int row0 = (tile >> 1) * 16;        // 256 row tiles
  const int n0   = (tile & 1) * 64;         // 2 col groups of 64

  v8f acc[4] = {};
  const float* ap = h + (size_t)(row0 + l16) * N1 + 2 * half;
  const float* bp = W2 + (size_t)(2 * half) * Psz + n0 + l16;
#pragma unroll 2
  for (int k0 = 0; k0 < N1; k0 += 4) {
    v2f a;
    a.x = ap[0];
    a.y = ap[1];
#pragma unroll
    for (int g = 0; g < 4; ++g) {
      v2f bb;
      bb.x = bp[g * 16];
      bb.y = bp[Psz + g * 16];
      acc[g] = __builtin_amdgcn_wmma_f32_16x16x4_f32(
          false, a, false, bb, (short)0, acc[g], false, false);
    }
    ap += 4;
    bp += 4 * Psz;
  }
#pragma unroll
  for (int g = 0; g < 4; ++g) {
#pragma unroll
    for (int j = 0; j < 8; ++j) {
      const int m = row0 + j + 8 * half;
      const int n = n0 + g * 16 + l16;
      y[(size_t)m * Psz + n] = acc[g][j] + b2[n];
    }
  }
}

// ---------------------------------------------------------------------------
// Kernel 5: LayerNorm over last dim (Psz=128), out = (y-mu)/sqrt(var+eps)*g+b
// grid = ROWS blocks, block = Psz threads (4 waves)
// ---------------------------------------------------------------------------
__global__ void pe_layernorm(const float* __restrict__ y,
                             const float* __restrict__ ln_g,
                             const float* __restrict__ ln_b,
                             float* __restrict__ out) {
  const int row = blockIdx.x;
  const int p   = threadIdx.x;   // 0..127
  __shared__ float s1[128];
  __shared__ float s2[128];
  const float v = y[(size_t)row * Psz + p];
  s1[p] = v;
  s2[p] = v * v;
  __syncthreads();
  for (int s = 64; s > 0; s >>= 1) {
    if (p < s) { s1[p] += s1[p + s]; s2[p] += s2[p + s]; }
    __syncthreads();
  }
  const float mu  = s1[0] * (1.0f / Psz);
  const float var = s2[0] * (1.0f / Psz) - mu * mu;   // population variance
  const float r   = rsqrtf(var + 1e-5f);
  out[(size_t)row * Psz + p] = (v - mu) * r * ln_g[p] + ln_b[p];
}

// ---------------------------------------------------------------------------
// Launch
// ---------------------------------------------------------------------------
extern "C" void kernel_launch(void* const* d_in, const int* in_sizes, int n_in,
                              void* d_out, int out_size, void* d_ws, size_t ws_size,
                              hipStream_t stream) {
  const float*     evidence   = (const float*)d_in[0];      // (B,T,E) f32
  const long long* boundaries = (const long long*)d_in[1];  // (B,K,2) i64
  const int*       slot_mask  = (const int*)d_in[2];        // (B,K)   i32
  const float*     W1         = (const float*)d_in[3];      // (STAT, 2P)
  const float*     b1         = (const float*)d_in[4];      // (2P,)
  const float*     W2         = (const float*)d_in[5];      // (2P, P)
  const float*     b2         = (const float*)d_in[6];      // (P,)
  const float*     ln_g       = (const float*)d_in[7];      // (P,)
  const float*     ln_b       = (const float*)d_in[8];      // (P,)
  float*           out        = (float*)d_out;              // (B,K,P) f32

  // workspace layout (floats)
  float* featsP = (float*)d_ws;                         // ROWS * KP
  float* W1p    = featsP + (size_t)ROWS * KP;           // KP * N1
  float* h      = W1p    + (size_t)KP * N1;             // ROWS * N1
  float* y      = h      + (size_t)ROWS * N1;           // ROWS * Psz
  // total: 3,870,720 floats (~15.5 MB)

  pe_pad_w1<<<KP, N1, 0, stream>>>(W1, W1p);
  pe_feats<<<ROWS, Esz, 0, stream>>>(evidence, boundaries, slot_mask, featsP);
  pe_gemm1_gelu<<<(ROWS / 16) * (N1 / 64) / 8, 256, 0, stream>>>(featsP, W1p, b1, h);
  pe_gemm2<<<(ROWS / 16) * (Psz / 64) / 8, 256, 0, stream>>>(h, W2, b2, y);
  pe_layernorm<<<ROWS, Psz, 0, stream>>>(y, ln_g, ln_b, out);
}